// Attention_4191888080922
// MI455X (gfx1250) — compile-verified
//
#include <hip/hip_runtime.h>
#include <math.h>

#define H2 512
#define NB 64
#define TK 2048
#define TSPLIT 8

typedef __attribute__((ext_vector_type(2))) float v2f;
typedef __attribute__((ext_vector_type(8))) float v8f;

#if __has_builtin(__builtin_amdgcn_tanhf)
__device__ __forceinline__ float fast_tanh(float x) { return __builtin_amdgcn_tanhf(x); }
#else
__device__ __forceinline__ float fast_tanh(float x) { return tanhf(x); }
#endif

// ---------------------------------------------------------------------------
// Kernel 1: dec_fea[b,o] = sum_k s_t_hat[b,k] * W_dp[o,k] + b_dp[o]
// M=64, N=512, K=512 GEMM via V_WMMA_F32_16X16X4_F32. One wave per 16x16 tile.
// (verified in round 1: lowers to pipelined v_wmma_f32_16x16x4_f32)
// ---------------------------------------------------------------------------
__global__ __launch_bounds__(32) void k_decproj_wmma(const float* __restrict__ S,
                                                     const float* __restrict__ W,
                                                     const float* __restrict__ bias,
                                                     float* __restrict__ dec) {
    const int lane = threadIdx.x;
    const int half = lane >> 4;
    const int l    = lane & 15;
    const int mtile = blockIdx.x & 3;   // 64/16  M tiles
    const int ntile = blockIdx.x >> 2;  // 512/16 N tiles
    const int m0 = mtile * 16;
    const int n0 = ntile * 16;

    v8f c = {};
    for (int k0 = 0; k0 < H2; k0 += 4) {
        const int kk = k0 + 2 * half;
        v2f a = *(const v2f*)(S + (m0 + l) * H2 + kk);   // s_t_hat[m, kk:kk+2]
        v2f b = *(const v2f*)(W + (n0 + l) * H2 + kk);   // W_dp[n, kk:kk+2] == B[kk,n]
        c = __builtin_amdgcn_wmma_f32_16x16x4_f32(false, a, false, b,
                                                  (short)0, c, false, false);
    }
    const int col = n0 + l;
    const float bv = bias[col];
#pragma unroll
    for (int j = 0; j < 8; ++j) {
        const int row = m0 + j + 8 * half;
        dec[row * H2 + col] = c[j] + bv;
    }
}

// ---------------------------------------------------------------------------
// Kernel 2: scores[b,t] = v_b + sum_n v_w[n]*tanh(EF[b,t,n] + dec[b,n] + cov[b,t]*W_c[n])
// 8 waves/block, one wave per t. Each lane owns 16 contiguous n, read as 4x b128
// (lane stride 64B -> full cacheline coalescing over the 2KB row). Row state in LDS.
// ---------------------------------------------------------------------------
__global__ __launch_bounds__(256) void k_scores(const float* __restrict__ EF,
                                                const float* __restrict__ dec,
                                                const float* __restrict__ cov,
                                                const float* __restrict__ Wc,
                                                const float* __restrict__ vw,
                                                const float* __restrict__ vb,
                                                float* __restrict__ scores) {
    __shared__ alignas(16) float s_dec[H2];
    __shared__ alignas(16) float s_wc[H2];
    __shared__ alignas(16) float s_vw[H2];
    const int b     = blockIdx.x >> 8;   // 256 t-chunks (of 8) per batch row
    const int chunk = blockIdx.x & 255;
    const int tid   = threadIdx.x;
    for (int i = tid; i < H2; i += 256) {
        s_dec[i] = dec[b * H2 + i];
        s_wc[i]  = Wc[i];
        s_vw[i]  = vw[i];
    }
    __syncthreads();

    const int wave = tid >> 5;
    const int lane = tid & 31;
    const int t    = chunk * 8 + wave;
    const float cv = cov[b * TK + t];
    const float* ef = EF + ((size_t)b * TK + t) * H2;

    float acc = 0.f;
    const int nb = lane * 16;
#pragma unroll
    for (int j = 0; j < 4; ++j) {
        const int n = nb + j * 4;
        const float4 e = *(const float4*)(ef + n);
        const float4 d = *(const float4*)(s_dec + n);
        const float4 w = *(const float4*)(s_wc + n);
        const float4 v = *(const float4*)(s_vw + n);
        acc += v.x * fast_tanh(e.x + d.x + cv * w.x);
        acc += v.y * fast_tanh(e.y + d.y + cv * w.y);
        acc += v.z * fast_tanh(e.z + d.z + cv * w.z);
        acc += v.w * fast_tanh(e.w + d.w + cv * w.w);
    }
#pragma unroll
    for (int off = 16; off > 0; off >>= 1)
        acc += __shfl_xor(acc, off, 32);
    if (lane == 0)
        scores[b * TK + t] = acc + vb[0];
}

// ---------------------------------------------------------------------------
// Kernel 3: masked softmax over t (single normalization; the /sum1 cancels in
// the renorm), plus coverage_new = coverage + attn. One block per batch row.
// ---------------------------------------------------------------------------
__global__ __launch_bounds__(256) void k_softmax(const float* __restrict__ scores,
                                                 const float* __restrict__ mask,
                                                 const float* __restrict__ cov,
                                                 float* __restrict__ attn,
                                                 float* __restrict__ cov_new) {
    __shared__ float red[8];
    const int b    = blockIdx.x;
    const int tid  = threadIdx.x;
    const int lane = tid & 31;
    const int wave = tid >> 5;
    const float* s = scores + b * TK;

    float m = -1e30f;
    for (int t = tid; t < TK; t += 256) m = fmaxf(m, s[t]);
#pragma unroll
    for (int off = 16; off > 0; off >>= 1) m = fmaxf(m, __shfl_xor(m, off, 32));
    if (lane == 0) red[wave] = m;
    __syncthreads();
    float m_all = red[0];
#pragma unroll
    for (int i = 1; i < 8; ++i) m_all = fmaxf(m_all, red[i]);
    __syncthreads();

    float sum = 0.f;
    for (int t = tid; t < TK; t += 256)
        sum += __expf(s[t] - m_all) * mask[b * TK + t];
#pragma unroll
    for (int off = 16; off > 0; off >>= 1) sum += __shfl_xor(sum, off, 32);
    if (lane == 0) red[wave] = sum;
    __syncthreads();
    float sum_all = 0.f;
#pragma unroll
    for (int i = 0; i < 8; ++i) sum_all += red[i];
    const float inv = 1.f / sum_all;

    for (int t = tid; t < TK; t += 256) {
        const float a = __expf(s[t] - m_all) * mask[b * TK + t] * inv;
        attn[b * TK + t]    = a;
        cov_new[b * TK + t] = cov[b * TK + t] + a;
    }
}

// ---------------------------------------------------------------------------
// Kernel 4a: partial context: part[b,ts,n] = sum_{t in slice ts} attn[b,t]*EO[b,t,n]
// t split 8 ways for occupancy (512 blocks, 4096 waves). float2 per thread,
// attn slice staged once in LDS. Deterministic fixed-order accumulation.
// ---------------------------------------------------------------------------
__global__ __launch_bounds__(256) void k_context_partial(const float* __restrict__ EO,
                                                         const float* __restrict__ attn,
                                                         float* __restrict__ part) {
    __shared__ float s_a[TK / TSPLIT];
    const int b   = blockIdx.x / TSPLIT;
    const int ts  = blockIdx.x % TSPLIT;
    const int tid = threadIdx.x;
    const int tlen = TK / TSPLIT;          // 256
    const int t0   = ts * tlen;
    const float* a  = attn + b * TK + t0;
    const float* eo = EO + ((size_t)b * TK + t0) * H2;

    s_a[tid] = a[tid];
    __syncthreads();

    const int n = tid * 2;
    float accx = 0.f, accy = 0.f;
    for (int i = 0; i < tlen; ++i) {
        const float av = s_a[i];
        const float2 e = *(const float2*)(eo + (size_t)i * H2 + n);
        accx += av * e.x;
        accy += av * e.y;
    }
    float2 r; r.x = accx; r.y = accy;
    *(float2*)(part + ((size_t)b * TSPLIT + ts) * H2 + n) = r;
}

// ---------------------------------------------------------------------------
// Kernel 4b: c_t[b,n] = sum_ts part[b,ts,n]  (fixed-order -> deterministic)
// ---------------------------------------------------------------------------
__global__ __launch_bounds__(256) void k_context_reduce(const float* __restrict__ part,
                                                        float* __restrict__ ct) {
    const int b = blockIdx.x;
    const int n = threadIdx.x * 2;
    float accx = 0.f, accy = 0.f;
#pragma unroll
    for (int ts = 0; ts < TSPLIT; ++ts) {
        const float2 p = *(const float2*)(part + ((size_t)b * TSPLIT + ts) * H2 + n);
        accx += p.x;
        accy += p.y;
    }
    float2 r; r.x = accx; r.y = accy;
    *(float2*)(ct + b * H2 + n) = r;
}

// ---------------------------------------------------------------------------
extern "C" void kernel_launch(void* const* d_in, const int* in_sizes, int n_in,
                              void* d_out, int out_size, void* d_ws, size_t ws_size,
                              hipStream_t stream) {
    const float* s_t_hat  = (const float*)d_in[0];
    const float* enc_out  = (const float*)d_in[1];
    const float* enc_feat = (const float*)d_in[2];
    const float* mask     = (const float*)d_in[3];
    const float* cov      = (const float*)d_in[4];
    const float* W_dp     = (const float*)d_in[5];
    const float* b_dp     = (const float*)d_in[6];
    const float* W_c      = (const float*)d_in[7];
    const float* v_w      = (const float*)d_in[8];
    const float* v_b      = (const float*)d_in[9];

    float* out     = (float*)d_out;
    float* ct      = out;                 // [64, 512]
    float* attn    = ct + NB * H2;        // [64, 2048]
    float* cov_new = attn + NB * TK;      // [64, 2048]

    float* dec    = (float*)d_ws;         // [64, 512]
    float* scores = dec + NB * H2;        // [64, 2048]
    float* part   = scores + NB * TK;     // [64, 8, 512]

    k_decproj_wmma<<<(NB / 16) * (H2 / 16), 32, 0, stream>>>(s_t_hat, W_dp, b_dp, dec);
    k_scores<<<NB * (TK / 8), 256, 0, stream>>>(enc_feat, dec, cov, W_c, v_w, v_b, scores);
    k_softmax<<<NB, 256, 0, stream>>>(scores, mask, cov, attn, cov_new);
    k_context_partial<<<NB * TSPLIT, 256, 0, stream>>>(enc_out, attn, part);
    k_context_reduce<<<NB, 256, 0, stream>>>(part, ct);
}